// CombinedLoss_420906795326
// MI455X (gfx1250) — compile-verified
//
#include <hip/hip_runtime.h>
#include <hip/hip_bf16.h>
#include <math.h>

// CDNA5 / gfx1250: wave32, WMMA fp32 16x16x4 for the Gram matrices.
typedef __attribute__((ext_vector_type(2))) float v2f;
typedef __attribute__((ext_vector_type(8))) float v8f;

#define NPTS 4096
#define DIMS 32
#define LOG_N 8.317766166719343f   /* log(4096) */
#define EPS_FIN 0.0025f            /* blur^2 = 0.05^2 */
#define N_STEPS 30
#define ROW_SPLIT 16               /* reduction-dim split for occupancy */
#define ROWS_PER_SPLIT (NPTS / ROW_SPLIT)  /* 256 */

// ---------------------------------------------------------------------------
// squared norms of each row
// ---------------------------------------------------------------------------
__global__ void norms_kernel(const float* __restrict__ X, float* __restrict__ out) {
    int i = blockIdx.x * blockDim.x + threadIdx.x;
    if (i >= NPTS) return;
    const float* row = X + (size_t)i * DIMS;
    float s = 0.f;
#pragma unroll
    for (int k = 0; k < DIMS; ++k) s += row[k] * row[k];
    out[i] = s;
}

// ---------------------------------------------------------------------------
// Cost matrix via fp32 WMMA:  C[i][j] = max(0.5*|a_i|^2 + 0.5*|b_j|^2 - a.b, 0)
// One wave computes one 16x16 tile; 8 chained V_WMMA_F32_16X16X4_F32 over K=32.
// ---------------------------------------------------------------------------
__global__ void gemm_cost_kernel(const float* __restrict__ A, const float* __restrict__ B,
                                 const float* __restrict__ an, const float* __restrict__ bn,
                                 float* __restrict__ Cost) {
    const int lane = threadIdx.x & 31;
    const int wave = threadIdx.x >> 5;
    const int I = (blockIdx.x * 8 + wave) * 16;
    const int J = blockIdx.y * 16;
    const int mr = lane & 15;
    const int kh = (lane >> 4) * 2;

    const float* arow = A + (size_t)(I + mr) * DIMS + kh;
    const float* brow = B + (size_t)(J + mr) * DIMS + kh;

    v8f acc = {0.f, 0.f, 0.f, 0.f, 0.f, 0.f, 0.f, 0.f};
#pragma unroll
    for (int kk = 0; kk < DIMS; kk += 4) {
        v2f a = *(const v2f*)(arow + kk);
        v2f b = *(const v2f*)(brow + kk);
        acc = __builtin_amdgcn_wmma_f32_16x16x4_f32(
            /*neg_a=*/false, a, /*neg_b=*/false, b,
            /*c_mod=*/(short)0, acc, /*reuse_a=*/false, /*reuse_b=*/false);
    }

    const int col = J + (lane & 15);
    const int rbase = I + ((lane >> 4) ? 8 : 0);
    const float hbn = 0.5f * bn[col];
#pragma unroll
    for (int v = 0; v < 8; ++v) {
        int row = rbase + v;
        float val = fmaxf(0.5f * an[row] + hbn - acc[v], 0.0f);
        Cost[(size_t)row * NPTS + col] = val;
    }
}

// ---------------------------------------------------------------------------
// Stage-1 reduction: per-block partial sum of sqrt(max(2*C,1e-12)) (energy)
// and partial max of C (diameter). float4 loads, 4096 elements per block.
// ---------------------------------------------------------------------------
__global__ void reduce_stage1_kernel(const float* __restrict__ M,
                                     float* __restrict__ psum,
                                     float* __restrict__ pmax, int do_max) {
    __shared__ float ssum[256];
    __shared__ float smax[256];
    const float4* Mv = (const float4*)(M + (size_t)blockIdx.x * 4096);
    float s = 0.f, mx = 0.f;
#pragma unroll
    for (int k = 0; k < 4; ++k) {
        float4 c = Mv[threadIdx.x + k * 256];
        s += sqrtf(fmaxf(2.0f * c.x, 1e-12f)) + sqrtf(fmaxf(2.0f * c.y, 1e-12f)) +
             sqrtf(fmaxf(2.0f * c.z, 1e-12f)) + sqrtf(fmaxf(2.0f * c.w, 1e-12f));
        mx = fmaxf(mx, fmaxf(fmaxf(c.x, c.y), fmaxf(c.z, c.w)));
    }
    ssum[threadIdx.x] = s;
    smax[threadIdx.x] = mx;
    __syncthreads();
    for (int off = 128; off > 0; off >>= 1) {
        if (threadIdx.x < off) {
            ssum[threadIdx.x] += ssum[threadIdx.x + off];
            smax[threadIdx.x] = fmaxf(smax[threadIdx.x], smax[threadIdx.x + off]);
        }
        __syncthreads();
    }
    if (threadIdx.x == 0) {
        psum[blockIdx.x] = ssum[0];
        if (do_max) pmax[blockIdx.x] = smax[0];
    }
}

// Stage-2: scalars[0]=diam, [1]=sum_dxy, [2]=sum_dxx, [3]=sum_dyy
__global__ void reduce_stage2_kernel(const float* __restrict__ psum_xy,
                                     const float* __restrict__ psum_xx,
                                     const float* __restrict__ psum_yy,
                                     const float* __restrict__ pmax_xy,
                                     float* __restrict__ scalars) {
    __shared__ float sh[256];
    const int tid = threadIdx.x;

    float mx = 0.f;
    for (int i = tid; i < 4096; i += 256) mx = fmaxf(mx, pmax_xy[i]);
    sh[tid] = mx; __syncthreads();
    for (int off = 128; off > 0; off >>= 1) {
        if (tid < off) sh[tid] = fmaxf(sh[tid], sh[tid + off]);
        __syncthreads();
    }
    if (tid == 0) scalars[0] = sh[0];
    __syncthreads();

    const float* srcs[3] = {psum_xy, psum_xx, psum_yy};
    for (int q = 0; q < 3; ++q) {
        float s = 0.f;
        for (int i = tid; i < 4096; i += 256) s += srcs[q][i];
        sh[tid] = s; __syncthreads();
        for (int off = 128; off > 0; off >>= 1) {
            if (tid < off) sh[tid] += sh[tid + off];
            __syncthreads();
        }
        if (tid == 0) scalars[1 + q] = sh[0];
        __syncthreads();
    }
}

__global__ void zero_kernel(float* __restrict__ p, int n) {
    int i = blockIdx.x * blockDim.x + threadIdx.x;
    if (i < n) p[i] = 0.f;
}

// ---------------------------------------------------------------------------
// Branchless online logsumexp update: exactly one v_exp_f32 per element.
// ---------------------------------------------------------------------------
__device__ __forceinline__ void lse_update(float& m, float& s, float tt) {
    float mn = fmaxf(m, tt);
    float e = __expf(fminf(m, tt) - mn);
    s = (tt > m) ? __fmaf_rn(s, e, 1.0f) : (s + e);
    m = mn;
}

__device__ __forceinline__ float sched_eps(const float* scalars, int t, int is_final) {
    if (is_final) return EPS_FIN;
    return fmaxf(scalars[0] * exp2f(-2.0f * (float)t), EPS_FIN);  // diam * 0.25^t
}

// ---------------------------------------------------------------------------
// Softmin partials. All 4 softmins are COLUMN reductions over row-major
// matrices: f<-C_yx, g<-C_xy, sx<-C_xx, sy<-C_yy.
// grid = (col_blocks=4, ROW_SPLIT=16, problems=4); block = 256 threads.
// Each thread owns 4 consecutive columns (b128 loads) over 256 rows and
// writes per-split (max, sum) partials.
// ---------------------------------------------------------------------------
__global__ void softmin_part_kernel(const float* __restrict__ Cxy, const float* __restrict__ Cyx,
                                    const float* __restrict__ Cxx, const float* __restrict__ Cyy,
                                    const float* __restrict__ fc, const float* __restrict__ gc,
                                    const float* __restrict__ sxc, const float* __restrict__ syc,
                                    float* __restrict__ pm, float* __restrict__ ps,
                                    const float* __restrict__ scalars, int t, int is_final) {
    __shared__ float vsh[ROWS_PER_SPLIT];

    const int prob = blockIdx.z;
    const float* M;
    const float* v;
    switch (prob) {
        case 0:  M = Cyx; v = gc;  break;
        case 1:  M = Cxy; v = fc;  break;
        case 2:  M = Cxx; v = sxc; break;
        default: M = Cyy; v = syc; break;
    }

    const int tid = threadIdx.x;
    const int r0 = blockIdx.y * ROWS_PER_SPLIT;
    vsh[tid] = v[r0 + tid];
    __syncthreads();

    const float inv_eps = 1.0f / sched_eps(scalars, t, is_final);
    const float neg_inv = -inv_eps;

    const int c0 = blockIdx.x * 1024 + tid * 4;
    const float* Mp = M + (size_t)r0 * NPTS + c0;

    float m0 = -3.0e38f, m1 = -3.0e38f, m2 = -3.0e38f, m3 = -3.0e38f;
    float s0 = 0.f, s1 = 0.f, s2 = 0.f, s3 = 0.f;

    for (int r = 0; r < ROWS_PER_SPLIT; ++r) {
        float4 c = *(const float4*)Mp;
        Mp += NPTS;
        float vrs = vsh[r] * inv_eps;
        lse_update(m0, s0, __fmaf_rn(c.x, neg_inv, vrs));
        lse_update(m1, s1, __fmaf_rn(c.y, neg_inv, vrs));
        lse_update(m2, s2, __fmaf_rn(c.z, neg_inv, vrs));
        lse_update(m3, s3, __fmaf_rn(c.w, neg_inv, vrs));
    }

    // partial layout: [(prob*ROW_SPLIT + rowsplit)*NPTS + col]
    const size_t pbase = ((size_t)prob * ROW_SPLIT + blockIdx.y) * NPTS + c0;
    float4 mv = {m0, m1, m2, m3};
    float4 sv = {s0, s1, s2, s3};
    *(float4*)(pm + pbase) = mv;
    *(float4*)(ps + pbase) = sv;
}

// ---------------------------------------------------------------------------
// Merge ROW_SPLIT partials per column, finish logsumexp, apply update rule.
//   annealing: out[c] = 0.5*(prev[c] + softmin)
//   final:     out[c] = softmin
// ---------------------------------------------------------------------------
__global__ void softmin_combine_kernel(const float* __restrict__ pm, const float* __restrict__ ps,
                                       const float* __restrict__ fc, const float* __restrict__ gc,
                                       const float* __restrict__ sxc, const float* __restrict__ syc,
                                       float* __restrict__ fn, float* __restrict__ gn,
                                       float* __restrict__ sxn, float* __restrict__ syn,
                                       const float* __restrict__ scalars, int t, int is_final) {
    const int idx = blockIdx.x * blockDim.x + threadIdx.x;  // 0 .. 4*NPTS-1
    const int prob = idx >> 12;
    const int col = idx & (NPTS - 1);

    const float* prev;
    float* out;
    switch (prob) {
        case 0:  prev = fc;  out = fn;  break;
        case 1:  prev = gc;  out = gn;  break;
        case 2:  prev = sxc; out = sxn; break;
        default: prev = syc; out = syn; break;
    }

    const size_t base = (size_t)prob * ROW_SPLIT * NPTS + col;
    float mj[ROW_SPLIT];
    float M = -3.0e38f;
#pragma unroll
    for (int j = 0; j < ROW_SPLIT; ++j) {
        mj[j] = pm[base + (size_t)j * NPTS];
        M = fmaxf(M, mj[j]);
    }
    float S = 0.f;
#pragma unroll
    for (int j = 0; j < ROW_SPLIT; ++j) {
        S += ps[base + (size_t)j * NPTS] * __expf(mj[j] - M);
    }

    const float eps = sched_eps(scalars, t, is_final);
    float lse = M + __logf(S);
    float sm = -eps * (lse - LOG_N);
    out[col] = is_final ? sm : 0.5f * (prev[col] + sm);
}

// ---------------------------------------------------------------------------
// Final scalar:  0.001 * sinkhorn + energy
// ---------------------------------------------------------------------------
__global__ void final_kernel(const float* __restrict__ ffin, const float* __restrict__ gfin,
                             const float* __restrict__ sxfin, const float* __restrict__ syfin,
                             const float* __restrict__ scalars, float* __restrict__ out) {
    __shared__ float sh[256];
    const int tid = threadIdx.x;
    float acc = 0.f;
    for (int i = tid; i < NPTS; i += 256)
        acc += ffin[i] + gfin[i] - sxfin[i] - syfin[i];
    sh[tid] = acc; __syncthreads();
    for (int off = 128; off > 0; off >>= 1) {
        if (tid < off) sh[tid] += sh[tid + off];
        __syncthreads();
    }
    if (tid == 0) {
        float sink = sh[0] / (float)NPTS;
        const float invNM = 1.0f / ((float)NPTS * (float)NPTS);
        float energy = (scalars[1] - 0.5f * scalars[2] - 0.5f * scalars[3]) * invNM;
        out[0] = 0.001f * sink + 1.0f * energy;
    }
}

// ---------------------------------------------------------------------------
// Orchestration. Workspace layout (floats), ~261 MB total:
//   Cxy, Cyx, Cxx, Cyy : 4 x 4096^2
//   xn, yn             : 2 x 4096
//   f/g/sx/sy ping-pong: 8 x 4096
//   f/g/sx/sy final    : 4 x 4096
//   energy partials    : 4 x 4096
//   scalars            : 4 (+pad)
//   softmin partials   : 2 x (4*ROW_SPLIT*4096)
// ---------------------------------------------------------------------------
extern "C" void kernel_launch(void* const* d_in, const int* in_sizes, int n_in,
                              void* d_out, int out_size, void* d_ws, size_t ws_size,
                              hipStream_t stream) {
    const float* x = (const float*)d_in[0];
    const float* y = (const float*)d_in[1];
    float* out = (float*)d_out;
    float* ws = (float*)d_ws;

    const size_t NM = (size_t)NPTS * (size_t)NPTS;
    float* Cxy = ws;
    float* Cyx = Cxy + NM;
    float* Cxx = Cyx + NM;
    float* Cyy = Cxx + NM;
    float* xn = Cyy + NM;
    float* yn = xn + NPTS;
    float* vecs = yn + NPTS;  // 8 * NPTS
    float* fb[2]  = {vecs + 0 * NPTS, vecs + 1 * NPTS};
    float* gb[2]  = {vecs + 2 * NPTS, vecs + 3 * NPTS};
    float* sxb[2] = {vecs + 4 * NPTS, vecs + 5 * NPTS};
    float* syb[2] = {vecs + 6 * NPTS, vecs + 7 * NPTS};
    float* fin = vecs + 8 * NPTS;  // 4 * NPTS
    float* psum_xy = fin + 4 * NPTS;
    float* psum_xx = psum_xy + 4096;
    float* psum_yy = psum_xx + 4096;
    float* pmax_xy = psum_yy + 4096;
    float* scalars = pmax_xy + 4096;
    float* pm = scalars + 16;                    // 4*ROW_SPLIT*NPTS
    float* ps = pm + 4 * ROW_SPLIT * NPTS;       // 4*ROW_SPLIT*NPTS

    // row norms
    norms_kernel<<<NPTS / 256, 256, 0, stream>>>(x, xn);
    norms_kernel<<<NPTS / 256, 256, 0, stream>>>(y, yn);

    // cost matrices via fp32 WMMA (Cyx = Cxy^T so every softmin reduces along
    // the strided row axis of a row-major matrix: a column reduction)
    dim3 ggrid(32, 256);
    gemm_cost_kernel<<<ggrid, 256, 0, stream>>>(x, y, xn, yn, Cxy);
    gemm_cost_kernel<<<ggrid, 256, 0, stream>>>(y, x, yn, xn, Cyx);
    gemm_cost_kernel<<<ggrid, 256, 0, stream>>>(x, x, xn, xn, Cxx);
    gemm_cost_kernel<<<ggrid, 256, 0, stream>>>(y, y, yn, yn, Cyy);

    // energy-distance sums + diameter
    reduce_stage1_kernel<<<4096, 256, 0, stream>>>(Cxy, psum_xy, pmax_xy, 1);
    reduce_stage1_kernel<<<4096, 256, 0, stream>>>(Cxx, psum_xx, pmax_xy, 0);
    reduce_stage1_kernel<<<4096, 256, 0, stream>>>(Cyy, psum_yy, pmax_xy, 0);
    reduce_stage2_kernel<<<1, 256, 0, stream>>>(psum_xy, psum_xx, psum_yy, pmax_xy, scalars);

    // init potentials to zero
    zero_kernel<<<(8 * NPTS + 255) / 256, 256, 0, stream>>>(vecs, 8 * NPTS);

    dim3 pgrid(NPTS / 1024, ROW_SPLIT, 4);  // (4, 16, 4)
    dim3 cgrid(4 * NPTS / 256);             // 64

    // eps-annealed Sinkhorn steps (ping-pong buffers)
    for (int t = 0; t < N_STEPS; ++t) {
        int c = t & 1, n = c ^ 1;
        softmin_part_kernel<<<pgrid, 256, 0, stream>>>(
            Cxy, Cyx, Cxx, Cyy,
            fb[c], gb[c], sxb[c], syb[c],
            pm, ps, scalars, t, 0);
        softmin_combine_kernel<<<cgrid, 256, 0, stream>>>(
            pm, ps,
            fb[c], gb[c], sxb[c], syb[c],
            fb[n], gb[n], sxb[n], syb[n],
            scalars, t, 0);
    }

    // final extrapolation at eps_fin (N_STEPS even -> converged state in buf 0)
    softmin_part_kernel<<<pgrid, 256, 0, stream>>>(
        Cxy, Cyx, Cxx, Cyy,
        fb[0], gb[0], sxb[0], syb[0],
        pm, ps, scalars, 0, 1);
    softmin_combine_kernel<<<cgrid, 256, 0, stream>>>(
        pm, ps,
        fb[0], gb[0], sxb[0], syb[0],
        fin, fin + NPTS, fin + 2 * NPTS, fin + 3 * NPTS,
        scalars, 0, 1);

    final_kernel<<<1, 256, 0, stream>>>(fin, fin + NPTS, fin + 2 * NPTS, fin + 3 * NPTS,
                                        scalars, out);
}